// krignn2_42348377539062
// MI455X (gfx1250) — compile-verified
//
#include <hip/hip_runtime.h>
#include <math.h>

// Problem constants (match reference)
#define B_    16
#define NK_   1024
#define NQ_   1024
#define DIN   3
#define DV    8
#define DH    128
#define NUGGET 1e-3f
#define EPS_  1e-12f

#define PB    64              // pivot block (panel) size
#define NSTEP (NK_ / PB)      // 16 elimination steps
#define NCOLS (NK_ + NQ_)     // 2048 augmented columns
#define NPB   (NCOLS / PB)    // 32 column blocks of width 64

#define LDST  68              // LDS row stride (words) for transposed B tile

typedef __attribute__((ext_vector_type(2))) float v2f;
typedef __attribute__((ext_vector_type(8))) float v8f;

// ---------------------------------------------------------------------------
// Cooperative staging: 64x32 global tile (row-major, ldb) -> transposed LDS
// BsT[n*LDST + k].  128 threads; global reads coalesced along n.
// ---------------------------------------------------------------------------
__device__ __forceinline__ void stage_B_T(const float* __restrict__ Bsrc, size_t ldb,
                                          float* __restrict__ BsT)
{
    const int t = threadIdx.x;          // 0..127
    const int n = t & 31;
#pragma unroll
    for (int it = 0; it < 16; ++it) {
        int k = (t >> 5) + it * 4;      // 0..63
        BsT[n * LDST + k] = Bsrc[(size_t)k * ldb + n];
    }
}

// ---------------------------------------------------------------------------
// 16(M) x 32(N) += 16x64 @ 64x32 with B read from transposed LDS tile.
// 32 x V_WMMA_F32_16X16X4_F32 per wave.
// Fragment layout (CDNA5 ISA 7.12.2):
//   A 16x4 : lane m = lane&15, reg r -> K = r + 2*(lane>>4)
//   B 4x16 : lane n = lane&15, reg r -> K = r + 2*(lane>>4)
//   C 16x16: reg r -> row r (lanes 0-15) / row r+8 (lanes 16-31), col lane&15
// A-row streamed as float4 chunks; B fragment pair = one ds_load_b64,
// conflict-free (LDST=68 -> each instruction touches all 64 banks once).
// ---------------------------------------------------------------------------
__device__ __forceinline__ void wmma_16x32_k64_lds(const float* __restrict__ A, size_t lda,
                                                   const float* __restrict__ BsT,
                                                   unsigned lane, v8f& acc0, v8f& acc1)
{
    const unsigned mn = lane & 15u;
    const unsigned g  = (lane >> 4) & 1u;
    const float4* arow = (const float4*)(A + (size_t)mn * lda);
#pragma unroll
    for (int kc = 0; kc < 4; ++kc) {               // K chunks of 16
        float4 ch[4] = {arow[kc * 4 + 0], arow[kc * 4 + 1],
                        arow[kc * 4 + 2], arow[kc * 4 + 3]};
#pragma unroll
        for (int kk = 0; kk < 4; ++kk) {           // 4-deep wmma steps
            float4 cc = ch[kk];
            v2f av, bv0, bv1;
            av[0] = g ? cc.z : cc.x;               // A[m][kc*16 + kk*4 + 2g]
            av[1] = g ? cc.w : cc.y;
            unsigned k = (unsigned)kc * 16u + (unsigned)kk * 4u + g * 2u;
            float2 b0 = *(const float2*)(BsT + mn * LDST + k);          // B[k..k+1][n]
            float2 b1 = *(const float2*)(BsT + (mn + 16u) * LDST + k);  // B[k..k+1][n+16]
            bv0[0] = b0.x; bv0[1] = b0.y;
            bv1[0] = b1.x; bv1[1] = b1.y;
            acc0 = __builtin_amdgcn_wmma_f32_16x16x4_f32(
                       false, av, false, bv0, (short)0, acc0, false, false);
            acc1 = __builtin_amdgcn_wmma_f32_16x16x4_f32(
                       false, av, false, bv1, (short)0, acc1, false, false);
        }
    }
}

// ---------------------------------------------------------------------------
// Fused MLP + elementwise scale: Y[r] = X[r] * mlp(X[r])   (in_f == out_f)
// ---------------------------------------------------------------------------
__global__ void mlp_scale_kernel(const float* __restrict__ X, float* __restrict__ Y,
                                 int rows, int in_f, int out_f,
                                 const float* __restrict__ w1, const float* __restrict__ b1,
                                 const float* __restrict__ w2, const float* __restrict__ b2,
                                 const float* __restrict__ w3, const float* __restrict__ b3)
{
    __shared__ float sh[64 * (DH + 1)];
    const int t = threadIdx.x;
    const int r = blockIdx.x * 64 + t;
    if (r >= rows) return;

    const float* x = X + (size_t)r * in_f;
    float xi[DV];
    for (int i = 0; i < in_f; ++i) xi[i] = x[i];

    float* h = &sh[t * (DH + 1)];
    for (int j = 0; j < DH; ++j) {
        float s = b1[j];
        for (int i = 0; i < in_f; ++i) s += xi[i] * w1[j * in_f + i];
        h[j] = fmaxf(s, 0.0f);
    }

    float o[DV];
    for (int v = 0; v < out_f; ++v) o[v] = b3[v];
    for (int j = 0; j < DH; ++j) {
        float s = b2[j];
        const float* w2r = w2 + (size_t)j * DH;
#pragma unroll 8
        for (int k = 0; k < DH; ++k) s += h[k] * w2r[k];
        s = fmaxf(s, 0.0f);
        for (int v = 0; v < out_f; ++v) o[v] += w3[v * DH + j] * s;
    }

    float* y = Y + (size_t)r * out_f;
    for (int v = 0; v < out_f; ++v) y[v] = xi[v] * o[v];
}

// ---------------------------------------------------------------------------
// Augmented matrix build: Aug[b][k][c] =
//   c <  NK : exp(-dist(Ks[b][k], Ks[b][c])) + (c==k)*NUGGET
//   c >= NK : exp(-dist(Qs[b][c-NK], Ks[b][k]))        (= C_qk^T)
// ---------------------------------------------------------------------------
__global__ void build_cov_kernel(const float* __restrict__ Ks,
                                 const float* __restrict__ Qs,
                                 float* __restrict__ Aug)
{
    const int b = blockIdx.y;
    const int k = blockIdx.x;
    const float* pk = Ks + ((size_t)b * NK_ + k) * DIN;
    const float a0 = pk[0], a1 = pk[1], a2 = pk[2];
    float* row = Aug + ((size_t)b * NK_ + k) * NCOLS;

    for (int c = threadIdx.x; c < NCOLS; c += blockDim.x) {
        const float* pc = (c < NK_)
            ? (Ks + ((size_t)b * NK_ + c) * DIN)
            : (Qs + ((size_t)b * NQ_ + (c - NK_)) * DIN);
        float d0 = a0 - pc[0], d1 = a1 - pc[1], d2 = a2 - pc[2];
        float d2s = d0 * d0 + d1 * d1 + d2 * d2;
        float dist = sqrtf(fmaxf(d2s, 0.0f) + EPS_);
        float v = __expf(-dist);
        if (c == k) v += NUGGET;
        row[c] = v;
    }
}

// ---------------------------------------------------------------------------
// Invert the 64x64 diagonal pivot block (Gauss-Jordan, SPD -> no pivoting).
// One block per batch; 64 threads; working [A|I] lives in LDS (33 KB).
// ---------------------------------------------------------------------------
__global__ void invert_diag64_kernel(const float* __restrict__ Aug,
                                     float* __restrict__ Winv, int j)
{
    __shared__ float M[PB][2 * PB + 1];
    const int b = blockIdx.x;
    const int t = threadIdx.x;

    const float* Ablk = Aug + ((size_t)b * NK_ + (size_t)j * PB) * NCOLS + (size_t)j * PB;
    for (int c = 0; c < PB; ++c) M[t][c] = Ablk[(size_t)t * NCOLS + c];
    for (int c = 0; c < PB; ++c) M[t][PB + c] = (c == t) ? 1.0f : 0.0f;
    __syncthreads();

    for (int p = 0; p < PB; ++p) {
        if (t == p) {
            float inv = 1.0f / M[p][p];
            for (int c = p; c < 2 * PB; ++c) M[p][c] *= inv;
        }
        __syncthreads();
        if (t != p) {
            float f = M[t][p];
            for (int c = p; c < 2 * PB; ++c) M[t][c] -= f * M[p][c];
        }
        __syncthreads();
    }

    float* w = Winv + (size_t)b * PB * PB;
    for (int c = 0; c < PB; ++c) w[(size_t)t * PB + c] = M[t][PB + c];
}

// ---------------------------------------------------------------------------
// Pivot panel scale: Prow = Winv(64x64) @ Aug[pivot rows] for columns > pivot.
// 128 threads = 4 waves sharing one staged B tile; wave w does rows w*16..+16.
// Grid: x = 32-wide col tiles, y = batch.
// ---------------------------------------------------------------------------
__global__ void gj_scale64_kernel(const float* __restrict__ Aug,
                                  const float* __restrict__ Winv,
                                  float* __restrict__ Prow, int j)
{
    __shared__ __align__(16) float BsT[32 * LDST];
    const int b = blockIdx.y;
    const size_t colbase = (size_t)(j + 1) * PB + (size_t)blockIdx.x * 32;

    const float* Bsrc = Aug + ((size_t)b * NK_ + (size_t)j * PB) * NCOLS + colbase;
    stage_B_T(Bsrc, NCOLS, BsT);
    __syncthreads();

    const unsigned w    = threadIdx.x >> 5;   // wave id = row subtile
    const unsigned lane = threadIdx.x & 31u;

    const float* W = Winv + (size_t)b * PB * PB + (size_t)w * 16 * PB;
    v8f a0 = {}, a1 = {};
    wmma_16x32_k64_lds(W, PB, BsT, lane, a0, a1);

    float* P = Prow + (size_t)b * PB * NCOLS + (size_t)w * 16 * NCOLS + colbase;
    const unsigned mn = lane & 15u;
    const unsigned mb = (lane >> 4) ? 8u : 0u;
#pragma unroll
    for (int r = 0; r < 8; ++r) {
        P[(size_t)(mb + r) * NCOLS + mn]      = a0[r];
        P[(size_t)(mb + r) * NCOLS + mn + 16] = a1[r];
    }
}

// ---------------------------------------------------------------------------
// Elimination update (FLOP-dominant): 128 threads = 4 waves per 64(row)x32(col)
// group, sharing one LDS-staged Prow tile.
//   non-pivot rows: Aug[i][cols] -= Aug[i][pivot cols] @ Prow
//   pivot rows    : Aug[pivot rows][cols] = Prow  (copy-back for later steps)
// Grid: x = 32-wide col tiles, y = 16 row groups of 64, z = batch.
// ---------------------------------------------------------------------------
__global__ void gj_update64_kernel(float* __restrict__ Aug,
                                   const float* __restrict__ Prow, int j)
{
    __shared__ __align__(16) float BsT[32 * LDST];
    const int b  = blockIdx.z;
    const int rg = blockIdx.y;                       // 64-row group
    const size_t colbase = (size_t)(j + 1) * PB + (size_t)blockIdx.x * 32;

    const float* P = Prow + (size_t)b * PB * NCOLS;
    float* Cg = Aug + ((size_t)b * NK_ + (size_t)rg * PB) * NCOLS + colbase;

    if (rg == j) {
        // pivot row group: copy scaled panel back into Aug (columns > pivot)
        const int t = threadIdx.x, n = t & 31;
        const float* Ps = P + colbase;
#pragma unroll
        for (int it = 0; it < 16; ++it) {
            int k = (t >> 5) + it * 4;
            Cg[(size_t)k * NCOLS + n] = Ps[(size_t)k * NCOLS + n];
        }
        return;
    }

    stage_B_T(P + colbase, NCOLS, BsT);
    __syncthreads();

    const unsigned w    = threadIdx.x >> 5;          // wave id = 16-row subtile
    const unsigned lane = threadIdx.x & 31u;

    const float* Apan = Aug + ((size_t)b * NK_ + (size_t)rg * PB + (size_t)w * 16) * NCOLS
                            + (size_t)j * PB;
    v8f a0 = {}, a1 = {};
    wmma_16x32_k64_lds(Apan, NCOLS, BsT, lane, a0, a1);

    float* Ct = Cg + (size_t)w * 16 * NCOLS;
    const unsigned mn = lane & 15u;
    const unsigned mb = (lane >> 4) ? 8u : 0u;
#pragma unroll
    for (int r = 0; r < 8; ++r) {
        size_t i0 = (size_t)(mb + r) * NCOLS + mn;
        Ct[i0]      -= a0[r];
        Ct[i0 + 16] -= a1[r];
    }
}

// ---------------------------------------------------------------------------
// Output contraction: out[b][q][v] = sum_k lam[b][k][q] * Vs[b][k][v]
// ---------------------------------------------------------------------------
__global__ void out_kernel(const float* __restrict__ Aug,
                           const float* __restrict__ Vs,
                           float* __restrict__ Out)
{
    const int b = blockIdx.y;
    const int q = blockIdx.x * blockDim.x + threadIdx.x;

    float acc[DV];
#pragma unroll
    for (int v = 0; v < DV; ++v) acc[v] = 0.0f;

    const float* lam = Aug + (size_t)b * NK_ * NCOLS + NK_ + q;
    const float* vsb = Vs + (size_t)b * NK_ * DV;
    for (int k = 0; k < NK_; ++k) {
        float l = lam[(size_t)k * NCOLS];
        const float* vv = vsb + (size_t)k * DV;
#pragma unroll
        for (int v = 0; v < DV; ++v) acc[v] += l * vv[v];
    }
    float* o = Out + ((size_t)b * NQ_ + q) * DV;
#pragma unroll
    for (int v = 0; v < DV; ++v) o[v] = acc[v];
}

// ---------------------------------------------------------------------------
extern "C" void kernel_launch(void* const* d_in, const int* in_sizes, int n_in,
                              void* d_out, int out_size, void* d_ws, size_t ws_size,
                              hipStream_t stream)
{
    const float* KEY   = (const float*)d_in[0];
    const float* VALUE = (const float*)d_in[1];
    const float* QUERY = (const float*)d_in[2];
    const float* W1w = (const float*)d_in[3],  *W1b = (const float*)d_in[4];
    const float* W2w = (const float*)d_in[5],  *W2b = (const float*)d_in[6];
    const float* W3w = (const float*)d_in[7],  *W3b = (const float*)d_in[8];
    const float* Wv1w = (const float*)d_in[9],  *Wv1b = (const float*)d_in[10];
    const float* Wv2w = (const float*)d_in[11], *Wv2b = (const float*)d_in[12];
    const float* Wv3w = (const float*)d_in[13], *Wv3b = (const float*)d_in[14];

    float* ws = (float*)d_ws;
    size_t off = 0;
    float* Ks   = ws + off; off += (size_t)B_ * NK_ * DIN;     // scaled keys
    float* Qs   = ws + off; off += (size_t)B_ * NQ_ * DIN;     // scaled queries
    float* Vs   = ws + off; off += (size_t)B_ * NK_ * DV;      // scaled values
    float* Winv = ws + off; off += (size_t)B_ * PB * PB;       // pivot-block inverses
    off = (off + 255) & ~(size_t)255;
    float* Prow = ws + off; off += (size_t)B_ * PB * NCOLS;    // scaled pivot panels
    off = (off + 255) & ~(size_t)255;
    float* Aug  = ws + off;                                    // [B, NK, 2048] augmented

    // 1) MLPs + elementwise scaling
    const int rowsK = B_ * NK_;
    mlp_scale_kernel<<<dim3((rowsK + 63) / 64), 64, 0, stream>>>(
        KEY, Ks, rowsK, DIN, DIN, W1w, W1b, W2w, W2b, W3w, W3b);
    mlp_scale_kernel<<<dim3((rowsK + 63) / 64), 64, 0, stream>>>(
        QUERY, Qs, rowsK, DIN, DIN, W1w, W1b, W2w, W2b, W3w, W3b);
    mlp_scale_kernel<<<dim3((rowsK + 63) / 64), 64, 0, stream>>>(
        VALUE, Vs, rowsK, DV, DV, Wv1w, Wv1b, Wv2w, Wv2b, Wv3w, Wv3b);

    // 2) Build augmented matrix [C_kk + nugget*I | C_qk^T] (L2-resident: 128 MB)
    build_cov_kernel<<<dim3(NK_, B_), 256, 0, stream>>>(Ks, Qs, Aug);

    // 3) Block Gauss-Jordan with 64-wide pivot panels (16 steps, SPD)
    for (int j = 0; j < NSTEP; ++j) {
        const int ctiles = 2 * (NPB - 1 - j);   // 32-wide column tiles right of pivot
        invert_diag64_kernel<<<dim3(B_), PB, 0, stream>>>(Aug, Winv, j);
        gj_scale64_kernel<<<dim3(ctiles, B_), 128, 0, stream>>>(Aug, Winv, Prow, j);
        gj_update64_kernel<<<dim3(ctiles, NK_ / PB, B_), 128, 0, stream>>>(Aug, Prow, j);
    }

    // 4) out[b,q,v] = sum_k lam[b,k,q] * Vs[b,k,v]
    out_kernel<<<dim3(NQ_ / 128, B_), 128, 0, stream>>>(Aug, Vs, (float*)d_out);
}